// SwavModel_37958920962068
// MI455X (gfx1250) — compile-verified
//
#include <hip/hip_runtime.h>
#include <math.h>

typedef __attribute__((ext_vector_type(2))) float v2f;
typedef __attribute__((ext_vector_type(8))) float v8f;

#define N_ROWS 32768
#define D_DIM  512
#define P_DIM  4096
#define EPSN   1e-12f
#define FMAXV  3.4028234663852886e38f
#define FMAXBITS 0x7F7FFFFFu

// ---------------------------------------------------------------------------
// Kernel 1: l2-normalize rows of x (wave per row); init rowmin bit array.
// ---------------------------------------------------------------------------
__global__ __launch_bounds__(256) void k_normalize(const float* __restrict__ x,
                                                   float* __restrict__ xn,
                                                   float* __restrict__ rn2,
                                                   unsigned* __restrict__ rowminbits) {
  int wid  = threadIdx.x >> 5;
  int lane = threadIdx.x & 31;
  int row  = blockIdx.x * 8 + wid;
  const float4* xr = (const float4*)(x + (size_t)row * D_DIM);
  float4 v[4];
  float ss = 0.f;
#pragma unroll
  for (int i = 0; i < 4; ++i) {
    v[i] = xr[lane + i * 32];
    ss += v[i].x * v[i].x + v[i].y * v[i].y + v[i].z * v[i].z + v[i].w * v[i].w;
  }
#pragma unroll
  for (int m = 16; m >= 1; m >>= 1) ss += __shfl_xor(ss, m, 32);
  float nrm   = sqrtf(ss);
  float scale = 1.0f / fmaxf(nrm, EPSN);
  if (lane == 0) {
    rn2[row]        = (nrm * scale) * (nrm * scale);  // == 1 normally
    rowminbits[row] = FMAXBITS;
  }
  float4* xo = (float4*)(xn + (size_t)row * D_DIM);
#pragma unroll
  for (int i = 0; i < 4; ++i) {
    float4 o;
    o.x = v[i].x * scale; o.y = v[i].y * scale;
    o.z = v[i].z * scale; o.w = v[i].w * scale;
    xo[lane + i * 32] = o;
  }
}

// ---------------------------------------------------------------------------
// Kernel 2: wn2[p] = ||W_p||^2 (wave per row); init colmin bit array.
// ---------------------------------------------------------------------------
__global__ __launch_bounds__(256) void k_wnorm(const float* __restrict__ W,
                                               float* __restrict__ wn2,
                                               unsigned* __restrict__ colminbits) {
  int wid  = threadIdx.x >> 5;
  int lane = threadIdx.x & 31;
  int row  = blockIdx.x * 8 + wid;
  const float4* wr = (const float4*)(W + (size_t)row * D_DIM);
  float ss = 0.f;
#pragma unroll
  for (int i = 0; i < 4; ++i) {
    float4 v = wr[lane + i * 32];
    ss += v.x * v.x + v.y * v.y + v.z * v.z + v.w * v.w;
  }
#pragma unroll
  for (int m = 16; m >= 1; m >>= 1) ss += __shfl_xor(ss, m, 32);
  if (lane == 0) {
    wn2[row]        = ss;
    colminbits[row] = FMAXBITS;
  }
}

// ---------------------------------------------------------------------------
// Kernel 3: logits = xn @ W^T via V_WMMA_F32_16X16X4_F32, with fused
// row/col d^2-min reductions in the epilogue.
// Block: 256 thr (8 waves), tile 128(M) x 128(P). Wave tile 32x64 (2x4 WMMA).
// LDS: double-buffered K-slices of 16, row stride 20 floats
//      (conflict-free 16-lane column reads; 80B rows keep float4 fills aligned).
// ---------------------------------------------------------------------------
#define BK  16
#define LDP 20
__global__ __launch_bounds__(256) void k_gemm(const float* __restrict__ xn,
                                              const float* __restrict__ W,
                                              const float* __restrict__ rn2,
                                              const float* __restrict__ wn2,
                                              float* __restrict__ logits,
                                              unsigned* __restrict__ rowminbits,
                                              unsigned* __restrict__ colminbits) {
  __shared__ alignas(16) float As[2][128][LDP];
  __shared__ alignas(16) float Bs[2][128][LDP];
  __shared__ float    rnl[128], wnl[128];
  __shared__ unsigned rminb[128], cminb[128];

  int tid  = threadIdx.x;
  int lane = tid & 31;
  int wid  = tid >> 5;
  int wm   = wid & 3;   // 4 waves along M  -> 32*wm
  int wn   = wid >> 2;  // 2 waves along P  -> 64*wn
  int r0   = blockIdx.y * 128;
  int c0   = blockIdx.x * 128;

  v8f acc[2][4];
#pragma unroll
  for (int mt = 0; mt < 2; ++mt)
#pragma unroll
    for (int nt = 0; nt < 4; ++nt) acc[mt][nt] = (v8f)(0.0f);

  int lrow = tid >> 2;        // 0..63 (row, +64 on second chunk)
  int lcol = (tid & 3) * 4;   // 0,4,8,12
  int hi   = lane >> 4;       // half-wave select
  int l15  = lane & 15;

  if (tid < 128) {
    rnl[tid]   = rn2[r0 + tid];
    wnl[tid]   = wn2[c0 + tid];
    rminb[tid] = FMAXBITS;
    cminb[tid] = FMAXBITS;
  }

  float4 pa[2], pb[2];
#pragma unroll
  for (int i = 0; i < 2; ++i) {
    pa[i] = *(const float4*)(xn + (size_t)(r0 + lrow + 64 * i) * D_DIM + lcol);
    pb[i] = *(const float4*)(W  + (size_t)(c0 + lrow + 64 * i) * D_DIM + lcol);
  }
#pragma unroll
  for (int i = 0; i < 2; ++i) {
    *(float4*)&As[0][lrow + 64 * i][lcol] = pa[i];
    *(float4*)&Bs[0][lrow + 64 * i][lcol] = pb[i];
  }
  __syncthreads();

  const int NIT = D_DIM / BK;  // 32
  for (int it = 0; it < NIT; ++it) {
    int buf = it & 1;
    if (it + 1 < NIT) {
      int kc = (it + 1) * BK;
#pragma unroll
      for (int i = 0; i < 2; ++i) {
        pa[i] = *(const float4*)(xn + (size_t)(r0 + lrow + 64 * i) * D_DIM + kc + lcol);
        pb[i] = *(const float4*)(W  + (size_t)(c0 + lrow + 64 * i) * D_DIM + kc + lcol);
      }
    }

#pragma unroll
    for (int kk = 0; kk < BK; kk += 4) {
      int kb = kk + hi * 2;  // ISA A/B frag: half-wave 0 -> K{0,1}, 1 -> K{2,3}
      v2f a[2], b[4];
#pragma unroll
      for (int mt = 0; mt < 2; ++mt) {
        a[mt].x = As[buf][wm * 32 + mt * 16 + l15][kb];
        a[mt].y = As[buf][wm * 32 + mt * 16 + l15][kb + 1];
      }
#pragma unroll
      for (int nt = 0; nt < 4; ++nt) {
        b[nt].x = Bs[buf][wn * 64 + nt * 16 + l15][kb];
        b[nt].y = Bs[buf][wn * 64 + nt * 16 + l15][kb + 1];
      }
#pragma unroll
      for (int mt = 0; mt < 2; ++mt)
#pragma unroll
        for (int nt = 0; nt < 4; ++nt)
          acc[mt][nt] = __builtin_amdgcn_wmma_f32_16x16x4_f32(
              false, a[mt], false, b[nt], (short)0, acc[mt][nt], false, false);
    }

    if (it + 1 < NIT) {
#pragma unroll
      for (int i = 0; i < 2; ++i) {
        *(float4*)&As[buf ^ 1][lrow + 64 * i][lcol] = pa[i];
        *(float4*)&Bs[buf ^ 1][lrow + 64 * i][lcol] = pb[i];
      }
    }
    __syncthreads();
  }

  // ---- store logits. C/D layout: VGPR v holds (M = v + 8*hi, N = l15) ----
#pragma unroll
  for (int mt = 0; mt < 2; ++mt)
#pragma unroll
    for (int nt = 0; nt < 4; ++nt) {
      int colg = c0 + wn * 64 + nt * 16 + l15;
#pragma unroll
      for (int v = 0; v < 8; ++v) {
        int rowg = r0 + wm * 32 + mt * 16 + v + 8 * hi;
        logits[(size_t)rowg * P_DIM + colg] = acc[mt][nt][v];
      }
    }

  // ---- fused min reductions: d^2 = rn2[r] + wn2[c] - 2*L ----
  float wv[4];
#pragma unroll
  for (int nt = 0; nt < 4; ++nt) wv[nt] = wnl[wn * 64 + nt * 16 + l15];
  float rv[2][8];
#pragma unroll
  for (int mt = 0; mt < 2; ++mt)
#pragma unroll
    for (int v = 0; v < 8; ++v) rv[mt][v] = rnl[wm * 32 + mt * 16 + v + 8 * hi];

  // row mins: one row lives in one VGPR across a 16-lane half
#pragma unroll
  for (int mt = 0; mt < 2; ++mt)
#pragma unroll
    for (int v = 0; v < 8; ++v) {
      float rp = FMAXV;
#pragma unroll
      for (int nt = 0; nt < 4; ++nt)
        rp = fminf(rp, wv[nt] - 2.0f * acc[mt][nt][v]);
      rp = fminf(rp, __shfl_xor(rp, 1, 32));
      rp = fminf(rp, __shfl_xor(rp, 2, 32));
      rp = fminf(rp, __shfl_xor(rp, 4, 32));
      rp = fminf(rp, __shfl_xor(rp, 8, 32));
      if (l15 == 0) {
        int rl = wm * 32 + mt * 16 + v + 8 * hi;
        float d2 = fmaxf(rv[mt][v] + rp, 0.0f);
        atomicMin(&rminb[rl], __float_as_uint(d2));
      }
    }

  // col mins: one column spans 8 VGPRs on a lane pair (l, l+16)
#pragma unroll
  for (int nt = 0; nt < 4; ++nt) {
    float cp = FMAXV;
#pragma unroll
    for (int mt = 0; mt < 2; ++mt)
#pragma unroll
      for (int v = 0; v < 8; ++v)
        cp = fminf(cp, rv[mt][v] - 2.0f * acc[mt][nt][v]);
    cp = fminf(cp, __shfl_xor(cp, 16, 32));
    if (hi == 0) {
      int cl = wn * 64 + nt * 16 + l15;
      float d2 = fmaxf(wv[nt] + cp, 0.0f);
      atomicMin(&cminb[cl], __float_as_uint(d2));
    }
  }
  __syncthreads();

  if (tid < 128)
    atomicMin(&rowminbits[r0 + tid], rminb[tid]);
  else
    atomicMin(&colminbits[c0 + tid - 128], cminb[tid - 128]);
}

// ---------------------------------------------------------------------------
// Kernel 4: final scalar reductions -> cvae_loss, prot_loss.
// Min arrays hold d^2 (clamped >= 0), sqrt here.
// ---------------------------------------------------------------------------
__global__ __launch_bounds__(256) void k_final(const unsigned* __restrict__ rowminbits,
                                               const unsigned* __restrict__ colminbits,
                                               const float* __restrict__ recon,
                                               const float* __restrict__ kl,
                                               const float* __restrict__ mmd,
                                               float* __restrict__ out2) {
  __shared__ float s1[256], s2[256];
  int tid = threadIdx.x;
  float a = 0.f, b = 0.f;
  for (int i = tid; i < N_ROWS; i += 256) a += sqrtf(__uint_as_float(rowminbits[i]));
  for (int i = tid; i < P_DIM; i += 256) b += sqrtf(__uint_as_float(colminbits[i]));
  s1[tid] = a; s2[tid] = b;
  __syncthreads();
  for (int s = 128; s > 0; s >>= 1) {
    if (tid < s) { s1[tid] += s1[tid + s]; s2[tid] += s2[tid + s]; }
    __syncthreads();
  }
  if (tid == 0) {
    out2[0] = recon[0] + 0.5f * kl[0] + mmd[0];                       // cvae
    out2[1] = 0.5f * (s1[0] / (float)N_ROWS) + 0.5f * (s2[0] / (float)P_DIM);
  }
}

// ---------------------------------------------------------------------------
extern "C" void kernel_launch(void* const* d_in, const int* in_sizes, int n_in,
                              void* d_out, int out_size, void* d_ws, size_t ws_size,
                              hipStream_t stream) {
  const float* x     = (const float*)d_in[0];
  const float* W     = (const float*)d_in[1];
  const float* recon = (const float*)d_in[2];
  const float* kl    = (const float*)d_in[3];
  const float* mmd   = (const float*)d_in[4];

  float* xn      = (float*)d_out;                       // [N, D]
  float* logits  = xn + (size_t)N_ROWS * D_DIM;         // [N, P]
  float* scalars = logits + (size_t)N_ROWS * P_DIM;     // [2]

  float* ws          = (float*)d_ws;
  float* rn2         = ws;                              // [N]
  float* wn2         = rn2 + N_ROWS;                    // [P]
  unsigned* rowbits  = (unsigned*)(wn2 + P_DIM);        // [N]
  unsigned* colbits  = rowbits + N_ROWS;                // [P]

  k_normalize<<<N_ROWS / 8, 256, 0, stream>>>(x, xn, rn2, rowbits);
  k_wnorm<<<P_DIM / 8, 256, 0, stream>>>(W, wn2, colbits);
  k_gemm<<<dim3(P_DIM / 128, N_ROWS / 128), 256, 0, stream>>>(
      xn, W, rn2, wn2, logits, rowbits, colbits);
  k_final<<<1, 256, 0, stream>>>(rowbits, colbits, recon, kl, mmd, scalars);
}